// SemiLinear_8040178778199
// MI455X (gfx1250) — compile-verified
//
#include <hip/hip_runtime.h>

#define BATCH 1024
#define IN_F  512
#define OUT_F 512

typedef float v2f __attribute__((ext_vector_type(2)));
typedef float v4f __attribute__((ext_vector_type(4)));
typedef float v8f __attribute__((ext_vector_type(8)));

// ---------------------------------------------------------------------------
// Kernel 1: proj[b,o] = sum_k x[b,k]*proj_w[o,k] + proj_b[o]   (written to out)
// One wave computes one 16x16 tile via V_WMMA_F32_16X16X4_F32 (full f32).
// ---------------------------------------------------------------------------
__global__ __launch_bounds__(32) void proj_wmma_kernel(
    const float* __restrict__ x, const float* __restrict__ pw,
    const float* __restrict__ pb, float* __restrict__ proj)
{
  const int lane = threadIdx.x;          // full wave, EXEC all ones
  const int row  = lane & 15;
  const int koff = (lane >> 4) << 1;     // 0 or 2
  const int bt   = blockIdx.x << 4;
  const int ot   = blockIdx.y << 4;

  const float* xr = x  + (size_t)(bt + row) * IN_F + koff;
  const float* wr = pw + (size_t)(ot + row) * IN_F + koff;

  v8f acc = {0.f, 0.f, 0.f, 0.f, 0.f, 0.f, 0.f, 0.f};
#pragma unroll 4
  for (int k = 0; k < IN_F; k += 4) {
    v2f a = *(const v2f*)(xr + k);
    v2f b = *(const v2f*)(wr + k);
    acc = __builtin_amdgcn_wmma_f32_16x16x4_f32(
        false, a, false, b, (short)0, acc, false, false);
  }

  const float bias  = pb[ot + row];
  const int   brow0 = bt + ((lane >> 4) << 3);
#pragma unroll
  for (int r = 0; r < 8; ++r) {
    proj[(size_t)(brow0 + r) * OUT_F + (ot + row)] = acc[r] + bias;
  }
}

// ---------------------------------------------------------------------------
// Kernel 2: AEG recurrence + sigmoid * proj, in-place on out.
// 64(batch) x 64(feature) tile per block, 4x4 register micro-tile per thread.
// K staged in chunks of 32 via GLOBAL_LOAD_ASYNC_TO_LDS_B32 (ASYNCcnt),
// double-buffered: chunk n+1's async loads are issued right after the
// barrier and overlap with chunk n's compute. The per-lane scatter of the
// async op performs the k-major transpose directly into LDS.
// ---------------------------------------------------------------------------
#define TB  64
#define TI  64
#define KC  32
#define PAD 4                  // row = 68 floats = 272B: 16B-aligned rows
#define NCHUNK (IN_F / KC)     // 16

__device__ __forceinline__ void async_b32_to_lds(const void* lds_ptr,
                                                 int gbyte_off,
                                                 const float* base) {
  unsigned lds_off = (unsigned)(uintptr_t)lds_ptr;   // low 32 bits = LDS offset
  asm volatile("global_load_async_to_lds_b32 %0, %1, %2"
               :: "v"(lds_off), "v"(gbyte_off), "s"(base)
               : "memory");
}

__global__ __launch_bounds__(256) void aeg_fuse_kernel(
    const float* __restrict__ x, const float* __restrict__ w,
    float* __restrict__ out /* holds proj on entry */)
{
  __shared__ float xT[2][KC][TB + PAD];   // xT[buf][kk][b_local]
  __shared__ float wT[2][KC][TI + PAD];   // wT[buf][kk][i_local]

  const int t  = threadIdx.x;
  const int b0 = blockIdx.x * TB;
  const int i0 = blockIdx.y * TI;

  const int tx = t & 15;   // batch micro-tile   -> b = b0 + 4*tx + d
  const int ty = t >> 4;   // feature micro-tile -> i = i0 + 4*ty + c

  const int skk  = t & (KC - 1);  // 0..31 : k within chunk (coalesced dim)
  const int srow = t >> 5;        // 0..7

  // Issue async staging of chunk at k0 into buffer `buf`.
  auto issue = [&](int k0, int buf) {
#pragma unroll
    for (int p = 0; p < TB; p += 8) {
      async_b32_to_lds(&xT[buf][skk][srow + p],
                       ((b0 + srow + p) * IN_F + k0 + skk) * 4, x);
      async_b32_to_lds(&wT[buf][skk][srow + p],
                       ((i0 + srow + p) * IN_F + k0 + skk) * 4, w);
    }
  };

  float r[4][4];
#pragma unroll
  for (int c = 0; c < 4; ++c)
#pragma unroll
    for (int d = 0; d < 4; ++d) r[c][d] = 0.f;

  issue(0, 0);

  for (int ch = 0; ch < NCHUNK; ++ch) {
    // Our chunk `ch` async stores must land, then all waves rendezvous.
    asm volatile("s_wait_asynccnt 0x0" ::: "memory");
    __syncthreads();
    // Overlap: start staging chunk ch+1 into the other buffer. Safe: every
    // wave finished reading that buffer (iteration ch-1) before this barrier.
    if (ch + 1 < NCHUNK) issue((ch + 1) * KC, (ch + 1) & 1);

    const int buf = ch & 1;
#pragma unroll
    for (int kk = 0; kk < KC; kk += 2) {
      v4f a0 = *(const v4f*)&wT[buf][kk    ][4 * ty];   // w[i][k]   (k even)
      v4f a1 = *(const v4f*)&wT[buf][kk + 1][4 * ty];   // w[i][k+1] (k odd)
      v4f b0 = *(const v4f*)&xT[buf][kk    ][4 * tx];   // x[b][k]
      v4f b1 = *(const v4f*)&xT[buf][kk + 1][4 * tx];   // x[b][k+1]
#pragma unroll
      for (int c = 0; c < 4; ++c) {
#pragma unroll
        for (int d = 0; d < 4; ++d) {
          if ((c & 1) == 0) {
            // even i: (i+k) even at even k -> opt1, then opt2 at odd k
            r[c][d] = (r[c][d] + a0[c]) * b0[d];
            r[c][d] = (r[c][d] + b1[d]) * a1[c];
          } else {
            // odd i: opt2 at even k, opt1 at odd k
            r[c][d] = (r[c][d] + b0[d]) * a0[c];
            r[c][d] = (r[c][d] + a1[c]) * b1[d];
          }
        }
      }
    }
  }

  // out = sigmoid(aeg) * proj  (1:1 element ownership, no races)
#pragma unroll
  for (int c = 0; c < 4; ++c) {
#pragma unroll
    for (int d = 0; d < 4; ++d) {
      const int row = b0 + 4 * tx + d;
      const int col = i0 + 4 * ty + c;
      const float proj = out[(size_t)row * OUT_F + col];
      const float sg   = 1.0f / (1.0f + __expf(-r[c][d]));
      out[(size_t)row * OUT_F + col] = sg * proj;
    }
  }
}

// ---------------------------------------------------------------------------
extern "C" void kernel_launch(void* const* d_in, const int* in_sizes, int n_in,
                              void* d_out, int out_size, void* d_ws, size_t ws_size,
                              hipStream_t stream) {
  (void)in_sizes; (void)n_in; (void)out_size; (void)d_ws; (void)ws_size;
  const float* x      = (const float*)d_in[0];   // (1024, 512)
  const float* weight = (const float*)d_in[1];   // (1, 512, 512) -> w[i*512+k]
  const float* proj_w = (const float*)d_in[2];   // (512, 512)
  const float* proj_b = (const float*)d_in[3];   // (512,)
  float* out = (float*)d_out;                    // (1024, 512)

  dim3 g1(BATCH / 16, OUT_F / 16);
  proj_wmma_kernel<<<g1, 32, 0, stream>>>(x, proj_w, proj_b, out);

  dim3 g2(BATCH / TB, OUT_F / TI);
  aeg_fuse_kernel<<<g2, 256, 0, stream>>>(x, weight, out);
}